// PearsonLoss_5145370820692
// MI455X (gfx1250) — compile-verified
//
#include <hip/hip_runtime.h>
#include <math.h>

typedef __attribute__((ext_vector_type(2))) float v2f;
typedef __attribute__((ext_vector_type(4))) float v4f;
typedef __attribute__((ext_vector_type(8))) float v8f;

#define T_DIM 16384
#define B_DIM 1024
#define ROWS_PER_BLOCK 16
#define WAVES_PER_BLOCK 8
#define GROUP_COLS 32        // 128 B per row per group (one cache line)

struct Acc { v8f p, t, pp, tt, pt; };
struct Grp { v4f p[4]; v4f t[4]; };

__device__ __forceinline__ void wmma5(Acc& a, v2f ap, v2f at, v2f ones) {
  a.p  = __builtin_amdgcn_wmma_f32_16x16x4_f32(false, ap, false, ones, (short)0, a.p,  false, false);
  a.t  = __builtin_amdgcn_wmma_f32_16x16x4_f32(false, at, false, ones, (short)0, a.t,  false, false);
  a.pp = __builtin_amdgcn_wmma_f32_16x16x4_f32(false, ap, false, ap,   (short)0, a.pp, false, false);
  a.tt = __builtin_amdgcn_wmma_f32_16x16x4_f32(false, at, false, at,   (short)0, a.tt, false, false);
  a.pt = __builtin_amdgcn_wmma_f32_16x16x4_f32(false, ap, false, at,   (short)0, a.pt, false, false);
}

// Load one 128B/row group: 4 x b128 per matrix per lane (contiguous 64B/lane).
__device__ __forceinline__ void load_group(const float* pB, const float* tB, Grp& g) {
#pragma unroll
  for (int j = 0; j < 4; ++j) {
    g.p[j] = *(const v4f*)(pB + 4 * j);
    g.t[j] = *(const v4f*)(tB + 4 * j);
  }
}

// Each v4f feeds two WMMA sub-chunks: halves are the lane's K{0,1} (lower
// half-wave) or K{2,3} (upper half-wave) operands.  Column->K mapping is a
// permutation of the 32 group columns; all five statistics are K-order
// independent, and P and T share the permutation, so diagonals stay exact.
__device__ __forceinline__ void process_group(Acc& a, const Grp& g, v2f ones) {
#pragma unroll
  for (int j = 0; j < 4; ++j) {
    v2f ap0 = {g.p[j][0], g.p[j][1]}, at0 = {g.t[j][0], g.t[j][1]};
    v2f ap1 = {g.p[j][2], g.p[j][3]}, at1 = {g.t[j][2], g.t[j][3]};
    wmma5(a, ap0, at0, ones);
    wmma5(a, ap1, at1, ones);
  }
}

__global__ __launch_bounds__(256) void pearson_partial_kernel(
    const float* __restrict__ pred, const float* __restrict__ target,
    float* __restrict__ partial, int splits)
{
  const int lane = threadIdx.x & 31;
  const int wave = threadIdx.x >> 5;
  const int sidx = blockIdx.y * WAVES_PER_BLOCK + wave;
  const int rowBase = blockIdx.x * ROWS_PER_BLOCK;
  const int colsPerSplit = T_DIM / splits;
  const int colBase = sidx * colsPerSplit;

  const int r16  = lane & 15;
  const int half = lane >> 4;                       // 0: cols [g,g+16), 1: [g+16,g+32)
  const size_t rowOfs = (size_t)(rowBase + r16) * T_DIM;

  const float* pB = pred   + rowOfs + colBase + half * 16;
  const float* tB = target + rowOfs + colBase + half * 16;

  Acc a;
  a.p = (v8f){}; a.t = (v8f){}; a.pp = (v8f){}; a.tt = (v8f){}; a.pt = (v8f){};
  const v2f ones = {1.0f, 1.0f};

  const int nGroups = colsPerSplit / GROUP_COLS;    // even for all split choices

  Grp b0, b1;
  load_group(pB, tB, b0);
  for (int g = 0; g < nGroups; g += 2) {
    // Pipeline: keep 8 B128 loads outstanding while 40 WMMAs retire.
    load_group(pB + (size_t)(g + 1) * GROUP_COLS, tB + (size_t)(g + 1) * GROUP_COLS, b1);
    __builtin_prefetch(pB + (size_t)(g + 4) * GROUP_COLS, 0, 3);
    __builtin_prefetch(tB + (size_t)(g + 4) * GROUP_COLS, 0, 3);
    process_group(a, b0, ones);

    if (g + 2 < nGroups)
      load_group(pB + (size_t)(g + 2) * GROUP_COLS, tB + (size_t)(g + 2) * GROUP_COLS, b0);
    __builtin_prefetch(pB + (size_t)(g + 5) * GROUP_COLS, 0, 3);
    __builtin_prefetch(tB + (size_t)(g + 5) * GROUP_COLS, 0, 3);
    process_group(a, b1, ones);
  }

  // Diagonal / row-sum extraction.  C/D 16x16 f32 layout:
  //   row m<8  : VGPR m,   lane m        (lanes 0-7)
  //   row m>=8 : VGPR m-8, lane 16+m     (lanes 24-31)
  if (lane < 8 || lane >= 24) {
    const int r    = (lane < 8) ? lane : (lane - 16);
    const int vidx = (lane < 8) ? lane : (lane - 24);
    float sp = 0.f, st = 0.f, spp = 0.f, stt = 0.f, spt = 0.f;
#pragma unroll
    for (int j = 0; j < 8; ++j) {
      if (vidx == j) { sp = a.p[j]; st = a.t[j]; spp = a.pp[j]; stt = a.tt[j]; spt = a.pt[j]; }
    }
    size_t base = ((size_t)sidx * B_DIM + (size_t)(rowBase + r)) * 5;
    partial[base + 0] = sp;
    partial[base + 1] = st;
    partial[base + 2] = spp;
    partial[base + 3] = stt;
    partial[base + 4] = spt;
  }
}

__global__ __launch_bounds__(256) void pearson_finalize_kernel(
    const float* __restrict__ partial, float* __restrict__ out, int splits)
{
  __shared__ float red[256];
  float local = 0.f;
  for (int row = threadIdx.x; row < B_DIM; row += 256) {
    float sp = 0.f, st = 0.f, spp = 0.f, stt = 0.f, spt = 0.f;
    for (int s = 0; s < splits; ++s) {
      size_t base = ((size_t)s * B_DIM + (size_t)row) * 5;
      sp  += partial[base + 0];
      st  += partial[base + 1];
      spp += partial[base + 2];
      stt += partial[base + 3];
      spt += partial[base + 4];
    }
    const float Tf = (float)T_DIM;
    float cp2 = spp - sp * sp / Tf;
    float ct2 = stt - st * st / Tf;
    float num = spt - sp * st / Tf;
    float var_p = cp2 / (Tf - 1.0f);
    float var_t = ct2 / (Tf - 1.0f);
    float denom = sqrtf(cp2 * ct2);
    float safe  = (denom > 0.f) ? denom : 1.0f;
    float corr  = num / safe;
    bool valid = (var_p > 1e-5f) && (var_t > 1e-5f) && (denom > 0.f) && !(corr != corr);
    local += valid ? (1.0f - corr) : 1.0f;
  }
  red[threadIdx.x] = local;
  __syncthreads();
  for (int s = 128; s > 0; s >>= 1) {
    if ((int)threadIdx.x < s) red[threadIdx.x] += red[threadIdx.x + s];
    __syncthreads();
  }
  if (threadIdx.x == 0) out[0] = red[0] / (float)B_DIM;
}

extern "C" void kernel_launch(void* const* d_in, const int* in_sizes, int n_in,
                              void* d_out, int out_size, void* d_ws, size_t ws_size,
                              hipStream_t stream) {
  const float* pred   = (const float*)d_in[0];
  const float* target = (const float*)d_in[1];
  float* out     = (float*)d_out;
  float* partial = (float*)d_ws;

  int gridY = 8;
  while (gridY > 1 &&
         (size_t)(gridY * WAVES_PER_BLOCK) * B_DIM * 5 * sizeof(float) > ws_size)
    gridY >>= 1;
  const int splits = gridY * WAVES_PER_BLOCK;

  dim3 grid(B_DIM / ROWS_PER_BLOCK, gridY);
  pearson_partial_kernel<<<grid, 256, 0, stream>>>(pred, target, partial, splits);
  pearson_finalize_kernel<<<1, 256, 0, stream>>>(partial, out, splits);
}